// GetYtr_85761906966757
// MI455X (gfx1250) — compile-verified
//
#include <hip/hip_runtime.h>

// ---------------------------------------------------------------------------
// GetYtr: per-batch radix-select of the R-th largest element, then binary
// mask emitted as complex64 (float pairs). B=8, L=64, H=W=256 per reference.
// ---------------------------------------------------------------------------

typedef __attribute__((ext_vector_type(4))) float v4f;
typedef __attribute__((ext_vector_type(4))) int   v4i;

#define BATCHES 8
#define MELEM   (64LL * 256 * 256)                 /* 4,194,304 per batch   */
#define RRANK   ((unsigned)((MELEM + 5) / 6))      /* ceil(M/6) = 699,051   */
#define NBINS   4096

#if defined(__gfx1250__) && \
    __has_builtin(__builtin_amdgcn_global_load_async_to_lds_b128) && \
    __has_builtin(__builtin_amdgcn_s_wait_asynccnt)
#define USE_ASYNC_LDS 1
#else
#define USE_ASYNC_LDS 0
#endif

// Monotone float -> uint key: preserves total order (and equality) of floats,
// so selecting / comparing in key space is exact vs. the reference.
__device__ __forceinline__ unsigned fkey(float f) {
  unsigned u = __float_as_uint(f);
  return u ^ (((unsigned)((int)u >> 31)) | 0x80000000u);
}

#if USE_ASYNC_LDS
typedef __attribute__((address_space(1))) v4i* gv4i_p;   // global int4*
typedef __attribute__((address_space(3))) v4i* lv4i_p;   // LDS int4*

__device__ __forceinline__ void async_copy16(const v4f* gsrc, v4f* ldst) {
  __builtin_amdgcn_global_load_async_to_lds_b128(
      (gv4i_p)gsrc, (lv4i_p)ldst, 0, 0);
}
#endif

// --- histogram pass (LDS histogram, async double-buffered tile stream) ----
// Defined first so the disassembly snippet shows this kernel.
__global__ void __launch_bounds__(256)
hist_kernel(const float* __restrict__ Y, unsigned* __restrict__ hist,
            const uint2* __restrict__ state,
            int shift, int prefixLen, int bins) {
  __shared__ unsigned lhist[NBINS];
#if USE_ASYNC_LDS
  __shared__ v4f tile[2][256];
#endif
  const int tid = threadIdx.x;
  const int b   = blockIdx.y;

  for (int i = tid; i < bins; i += blockDim.x) lhist[i] = 0u;
  __syncthreads();

  const unsigned prefix = state[b].x;         // scalar (uniform) load
  const int ps = 32 - prefixLen;
  const v4f* Yb = (const v4f*)(Y + (long long)b * MELEM);
  const long long nvec   = MELEM / 4;
  const long long stride = (long long)gridDim.x * blockDim.x;
  const long long v0     = (long long)blockIdx.x * blockDim.x + tid;
  const int iters = (int)(nvec / stride);     // grid sized for exact division

  auto accum = [&](float f) {
    unsigned k = fkey(f);
    if (prefixLen == 0 || (k >> ps) == prefix)
      atomicAdd(&lhist[(k >> shift) & (unsigned)(bins - 1)], 1u);
  };

#if USE_ASYNC_LDS
  if (iters > 0) async_copy16(Yb + v0, &tile[0][tid]);
  for (int s = 0; s + 1 < iters; ++s) {
    async_copy16(Yb + v0 + (long long)(s + 1) * stride, &tile[(s + 1) & 1][tid]);
    __builtin_amdgcn_s_wait_asynccnt(1);      // stage s landed in LDS
    asm volatile("" ::: "memory");
    v4f d = tile[s & 1][tid];
    accum(d.x); accum(d.y); accum(d.z); accum(d.w);
  }
  if (iters > 0) {
    __builtin_amdgcn_s_wait_asynccnt(0);
    asm volatile("" ::: "memory");
    v4f d = tile[(iters - 1) & 1][tid];
    accum(d.x); accum(d.y); accum(d.z); accum(d.w);
  }
#else
  for (int s = 0; s < iters; ++s) {
    v4f d = Yb[v0 + (long long)s * stride];   // global_load_b128
    accum(d.x); accum(d.y); accum(d.z); accum(d.w);
  }
#endif

  __syncthreads();
  unsigned* gh = hist + b * NBINS;
  for (int i = tid; i < bins; i += blockDim.x) {
    unsigned c = lhist[i];
    if (c) atomicAdd(&gh[i], c);
  }
}

// --- workspace init -------------------------------------------------------
__global__ void init_ws(unsigned* __restrict__ hist, uint2* __restrict__ state) {
  int i = blockIdx.x * blockDim.x + threadIdx.x;
  if (i < BATCHES * NBINS) hist[i] = 0u;
  if (i < BATCHES) state[i] = make_uint2(0u, RRANK);  // {prefix, rank}
}

__global__ void zero_hist(unsigned* __restrict__ hist) {
  int i = blockIdx.x * blockDim.x + threadIdx.x;
  if (i < BATCHES * NBINS) hist[i] = 0u;
}

// --- pick the bin holding the rank-th largest, refine prefix --------------
__global__ void select_kernel(const unsigned* __restrict__ hist,
                              uint2* __restrict__ state, int bins, int bits) {
  if (threadIdx.x != 0) return;
  const int b = blockIdx.x;
  const unsigned* h = hist + b * NBINS;
  const unsigned rank = state[b].y;
  unsigned long long cum = 0;
  int sel = 0;
  unsigned newRank = rank;
  for (int i = bins - 1; i >= 0; --i) {       // descending = from largest key
    unsigned long long c = h[i];
    if (cum + c >= (unsigned long long)rank) {
      sel = i;
      newRank = (unsigned)(rank - cum);
      break;
    }
    cum += c;
  }
  state[b].x = (state[b].x << bits) | (unsigned)sel;
  state[b].y = newRank;
}

// --- mask pass: complex64 {0|1, 0} with non-temporal b128 stores ----------
// Batch index comes from blockIdx.y: no 64-bit division in the hot loop,
// threshold key is a single scalar load per block.
__global__ void __launch_bounds__(256)
mask_kernel(const float* __restrict__ Y, float* __restrict__ out,
            const uint2* __restrict__ state) {
  const int b = blockIdx.y;
  const unsigned thr = state[b].x;            // full 32-bit key of threshold
  const v4f* Yv = (const v4f*)(Y + (long long)b * MELEM);
  v4f* Ov = (v4f*)(out + (long long)b * MELEM * 2);
  const long long nvec   = MELEM / 4;
  const long long stride = (long long)gridDim.x * blockDim.x;
  for (long long v = (long long)blockIdx.x * blockDim.x + threadIdx.x;
       v < nvec; v += stride) {
    v4f d = Yv[v];
    v4f o0, o1;
    o0.x = (fkey(d.x) >= thr) ? 1.0f : 0.0f;  o0.y = 0.0f;
    o0.z = (fkey(d.y) >= thr) ? 1.0f : 0.0f;  o0.w = 0.0f;
    o1.x = (fkey(d.z) >= thr) ? 1.0f : 0.0f;  o1.y = 0.0f;
    o1.z = (fkey(d.w) >= thr) ? 1.0f : 0.0f;  o1.w = 0.0f;
    __builtin_nontemporal_store(o0, &Ov[2 * v]);      // keep input resident in L2
    __builtin_nontemporal_store(o1, &Ov[2 * v + 1]);
  }
}

extern "C" void kernel_launch(void* const* d_in, const int* in_sizes, int n_in,
                              void* d_out, int out_size, void* d_ws, size_t ws_size,
                              hipStream_t stream) {
  const float* Y = (const float*)d_in[0];
  float* out = (float*)d_out;

  unsigned* hist = (unsigned*)d_ws;                                   // 128 KB
  uint2* state = (uint2*)((char*)d_ws + BATCHES * NBINS * sizeof(unsigned));

  const int zgrid = (BATCHES * NBINS + 255) / 256;
  dim3 hgrid(256, BATCHES), hblock(256);

  init_ws<<<zgrid, 256, 0, stream>>>(hist, state);

  // pass 1: key bits [31:20]
  hist_kernel<<<hgrid, hblock, 0, stream>>>(Y, hist, state, 20, 0, 4096);
  select_kernel<<<BATCHES, 32, 0, stream>>>(hist, state, 4096, 12);
  zero_hist<<<zgrid, 256, 0, stream>>>(hist);

  // pass 2: key bits [19:8]
  hist_kernel<<<hgrid, hblock, 0, stream>>>(Y, hist, state, 8, 12, 4096);
  select_kernel<<<BATCHES, 32, 0, stream>>>(hist, state, 4096, 12);
  zero_hist<<<zgrid, 256, 0, stream>>>(hist);

  // pass 3: key bits [7:0] -> state[b].x is the exact 32-bit threshold key
  hist_kernel<<<hgrid, hblock, 0, stream>>>(Y, hist, state, 0, 24, 256);
  select_kernel<<<BATCHES, 32, 0, stream>>>(hist, state, 256, 8);

  mask_kernel<<<dim3(1024, BATCHES), 256, 0, stream>>>(Y, out, state);
}